// PHLoss_84877143704065
// MI455X (gfx1250) — compile-verified
//
#include <hip/hip_runtime.h>
#include <stdint.h>

// ---------------- problem geometry ----------------
#define IMG_W   768
#define IMG_H   768
#define NPIX    (IMG_W * IMG_H)      // 589824
#define NIMG    32
#define STRIP_H 8
#define TILE    (IMG_W * STRIP_H)    // 6144 elements per strip
#define NSTRIP  (IMG_H / STRIP_H)    // 96 strips per image
#define NBOUND  (NSTRIP - 1)         // 95 seams per image
#define PH_INF  0x7FFFFFFF
#define THRESH  0.5f

// ---------------- TDM availability ----------------
#if defined(__HIP_DEVICE_COMPILE__) && __has_builtin(__builtin_amdgcn_tensor_load_to_lds) && __has_builtin(__builtin_amdgcn_s_wait_tensorcnt)
#define PH_HAVE_TDM 1
#else
#define PH_HAVE_TDM 0
#endif

typedef unsigned int tdm_u32x4 __attribute__((ext_vector_type(4)));
typedef int          tdm_i32x4 __attribute__((ext_vector_type(4)));
typedef int          tdm_i32x8 __attribute__((ext_vector_type(8)));

// ---------------- union-find primitives ----------------
// Labels always point to smaller indices (atomicMin linking), so find()
// strictly decreases and terminates even under concurrent updates.
__device__ __forceinline__ int ph_find(volatile int* lab, int x) {
    int p = lab[x];
    while (p != x) { x = p; p = lab[x]; }
    return x;
}

__device__ __forceinline__ void ph_union(int* lab, int a, int b) {
    volatile int* vl = (volatile int*)lab;
    int ra = ph_find(vl, a);
    int rb = ph_find(vl, b);
    while (ra != rb) {
        if (ra < rb) { int t = ra; ra = rb; rb = t; }   // ra = hi, rb = lo
        int old = atomicMin(&lab[ra], rb);
        if (old == ra) break;                           // linked hi -> lo
        ra = ph_find(vl, old);
        rb = ph_find(vl, rb);
    }
}

// ---------------- kernel 1: strip-local CCL in LDS ----------------
// One block per 768x8 strip. Stage the strip's fp32 data into LDS via the
// Tensor Data Mover, threshold, run union-find entirely in LDS, then write
// fully-flattened global labels (root = min global pixel index; PH_INF = bg).
__global__ void __launch_bounds__(512)
ph_local_ccl(const float* __restrict__ src, int* __restrict__ labels) {
    extern __shared__ char smem[];
    float* fbuf = (float*)smem;                              // TILE floats (24KB)
    int*   lab  = (int*)(smem + (size_t)TILE * sizeof(float)); // TILE ints (24KB)

    const int img   = blockIdx.x / NSTRIP;
    const int strip = blockIdx.x % NSTRIP;
    const int gbase = img * NPIX + strip * TILE;   // strip is contiguous in memory
    const float* tp = src + gbase;

#if PH_HAVE_TDM
    // Wave 0 issues one TDM descriptor: 1-D tile of TILE dwords -> LDS.
    if (threadIdx.x < 32) {
        uint64_t ga = (uint64_t)(uintptr_t)tp;
        uint32_t ldsoff = (uint32_t)(uintptr_t)fbuf;   // low 32 bits of flat LDS addr
        tdm_u32x4 g0;
        g0[0] = 1u;                                    // count=1, user descriptor
        g0[1] = ldsoff;                                // lds_addr
        g0[2] = (unsigned)(ga & 0xFFFFFFFFu);          // global_addr[31:0]
        g0[3] = (unsigned)((ga >> 32) & 0x01FFFFFFu) | (2u << 30); // addr[56:32], type=2
        tdm_i32x8 g1;
        g1[0] = 0x00020000;                            // data_size=2 (4 bytes)
        g1[1] = (int)((unsigned)TILE << 16);           // tensor_dim0[15:0]
        g1[2] = 0x00010000;                            // tensor_dim1 = 1
        g1[3] = (int)((unsigned)TILE << 16);           // tile_dim0 = TILE
        g1[4] = 1;                                     // tile_dim1 = 1
        g1[5] = TILE;                                  // tensor_dim0_stride
        g1[6] = (int)((unsigned)TILE << 16);           // tensor_dim1_stride[15:0]
        g1[7] = 0;
        tdm_i32x4 gz = (tdm_i32x4)0;
#if defined(__clang_major__) && __clang_major__ >= 23
        tdm_i32x8 gz8 = (tdm_i32x8)0;
        __builtin_amdgcn_tensor_load_to_lds(g0, g1, gz, gz, gz8, 0);
#else
        __builtin_amdgcn_tensor_load_to_lds(g0, g1, gz, gz, 0);
#endif
        __builtin_amdgcn_s_wait_tensorcnt(0);
    }
#else
    for (int q = threadIdx.x; q < TILE; q += blockDim.x) fbuf[q] = tp[q];
#endif
    __syncthreads();

    // threshold -> initial labels
    for (int q = threadIdx.x; q < TILE; q += blockDim.x)
        lab[q] = (fbuf[q] > THRESH) ? q : PH_INF;
    __syncthreads();

    // 4-connectivity unions within the strip (left + up neighbors)
    for (int q = threadIdx.x; q < TILE; q += blockDim.x) {
        if (lab[q] == PH_INF) continue;
        int c = q % IMG_W;
        if (c > 0      && lab[q - 1]     != PH_INF) ph_union(lab, q, q - 1);
        if (q >= IMG_W && lab[q - IMG_W] != PH_INF) ph_union(lab, q, q - IMG_W);
    }
    __syncthreads();

    // flatten to root and emit global labels
    volatile int* vl = (volatile int*)lab;
    for (int q = threadIdx.x; q < TILE; q += blockDim.x) {
        int l = lab[q];
        labels[gbase + q] = (l == PH_INF) ? PH_INF : (gbase + ph_find(vl, q));
    }
}

// ---------------- kernel 2: merge strip seams globally ----------------
__global__ void __launch_bounds__(256)
ph_merge(int* __restrict__ labels) {
    int tid = blockIdx.x * blockDim.x + threadIdx.x;
    const int per_img = NBOUND * IMG_W;            // 72960
    if (tid >= NIMG * per_img) return;
    int img = tid / per_img;
    int r   = tid % per_img;
    int b   = r / IMG_W;
    int c   = r % IMG_W;
    int row = (b + 1) * STRIP_H;
    int p   = img * NPIX + row * IMG_W + c;
    int up  = p - IMG_W;
    if (labels[p] != PH_INF && labels[up] != PH_INF)
        ph_union(labels, p, up);
}

// ---------------- kernel 3: count roots per image ----------------
__global__ void __launch_bounds__(256)
ph_count(const int* __restrict__ labels, int* __restrict__ counts) {
    const int PXB = 6144;                 // pixels per block
    const int BPI = NPIX / PXB;           // 96 blocks per image (exact)
    int img   = blockIdx.x / BPI;
    int chunk = blockIdx.x % BPI;
    int base  = img * NPIX + chunk * PXB;
    int cnt = 0;
    for (int i = threadIdx.x; i < PXB; i += blockDim.x) {
        int g = base + i;
        cnt += (labels[g] == g) ? 1 : 0;   // roots only (bg holds PH_INF != g)
    }
    __shared__ int red[256];
    red[threadIdx.x] = cnt;
    __syncthreads();
    for (int s = 128; s > 0; s >>= 1) {
        if (threadIdx.x < s) red[threadIdx.x] += red[threadIdx.x + s];
        __syncthreads();
    }
    if (threadIdx.x == 0) atomicAdd(&counts[img], red[0]);
}

// ---------------- helpers ----------------
__global__ void ph_zero(int* __restrict__ p, int n) {
    int i = blockIdx.x * blockDim.x + threadIdx.x;
    if (i < n) p[i] = 0;
}

__global__ void ph_final(const int* __restrict__ counts, float* __restrict__ out) {
    if (threadIdx.x == 0) {
        float s = 0.0f;
        for (int i = 0; i < NIMG; ++i) {
            float d = fabsf((float)counts[i] - (float)counts[NIMG + i]);
            s += sqrtf(d * 0.5f);
        }
        out[0] = s / (float)NIMG;
    }
}

// ---------------- launch ----------------
extern "C" void kernel_launch(void* const* d_in, const int* in_sizes, int n_in,
                              void* d_out, int out_size, void* d_ws, size_t ws_size,
                              hipStream_t stream) {
    (void)in_sizes; (void)n_in; (void)out_size; (void)ws_size;
    const float* pred   = (const float*)d_in[0];
    const float* target = (const float*)d_in[1];

    int* labels = (int*)d_ws;                                            // B*N ints (~75.5 MB)
    int* counts = (int*)((char*)d_ws + (size_t)NIMG * NPIX * sizeof(int)); // 64 ints

    const size_t lds_bytes   = (size_t)TILE * (sizeof(float) + sizeof(int)); // 48 KB
    const int    local_grid  = NIMG * NSTRIP;                                // 3072
    const int    merge_grid  = (NIMG * NBOUND * IMG_W + 255) / 256;
    const int    count_grid  = NIMG * (NPIX / 6144);                         // 3072

    ph_zero<<<1, 64, 0, stream>>>(counts, 2 * NIMG);

    // pass 1: pred
    ph_local_ccl<<<local_grid, 512, lds_bytes, stream>>>(pred, labels);
    ph_merge    <<<merge_grid, 256, 0, stream>>>(labels);
    ph_count    <<<count_grid, 256, 0, stream>>>(labels, counts);

    // pass 2: target (label buffer fully rewritten by ph_local_ccl)
    ph_local_ccl<<<local_grid, 512, lds_bytes, stream>>>(target, labels);
    ph_merge    <<<merge_grid, 256, 0, stream>>>(labels);
    ph_count    <<<count_grid, 256, 0, stream>>>(labels, counts + NIMG);

    ph_final<<<1, 32, 0, stream>>>(counts, (float*)d_out);
}